// SolveAll_9938554323597
// MI455X (gfx1250) — compile-verified
//
#include <hip/hip_runtime.h>
#include <math.h>

// ---------------------------------------------------------------------------
// Problem constants (match reference)
// ---------------------------------------------------------------------------
constexpr int NXQ = 40, NYQ = 40;
constexpr int NNODE  = NXQ * NYQ;            // 1600
constexpr int NDOF   = 2 * NNODE;            // 3200
constexpr int NELEM  = 2 * (NXQ-1)*(NYQ-1);  // 3042
constexpr int NDLOAD = 60;
constexpr int NFIX   = 80;
constexpr int NITER  = NDOF;                 // 3200 CG steps (reference scan length)
constexpr int NCHUNK = NDOF / 64;            // 50 wmma chunks per dot product
constexpr int NTRI   = 21;                   // unique entries of symmetric 6x6

// output layout (floats): u | uxcoord | uycoord | material | total
constexpr int OU = 0, OUX = 3200, OUY = 4800, OMAT = 6400, OTOT = 9442;

// ---------------------------------------------------------------------------
// Workspace layout (bytes)
// ---------------------------------------------------------------------------
constexpr size_t WS_RESID = 0;                                    // double[NDOF]
constexpr size_t WS_TOTAL = WS_RESID + (size_t)NDOF*8;            // double
constexpr size_t WS_MASK  = WS_TOTAL + 8;                         // char[NDOF]
constexpr size_t WS_FV    = ((WS_MASK + NDOF + 7)/8)*8;           // double[NDOF]
constexpr size_t WS_EST   = ((WS_FV + (size_t)NDOF*8 + 15)/16)*16;// float[NTRI*NELEM]
constexpr size_t WS_NOD   = WS_EST + (size_t)NTRI*NELEM*4;        // u16[3*NELEM]

// ---------------------------------------------------------------------------
// LDS layout for the persistent CG kernel (dynamic shared memory)
// est table + p + Kp + node table all LDS-resident: ~305 KB < 320 KB/WGP
// ---------------------------------------------------------------------------
constexpr unsigned L_EST  = 0;                              // float[NTRI*NELEM] 255528 B
constexpr unsigned L_P    = L_EST + NTRI*NELEM*4;           // double[NDOF]
constexpr unsigned L_KP   = L_P  + NDOF*8;                  // float[NDOF]
constexpr unsigned L_NOD  = L_KP + NDOF*4;                  // u16[3*NELEM]
constexpr unsigned L_SCR  = ((L_NOD + 3*NELEM*2 + 7)/8)*8;  // double[32] wave partials
constexpr unsigned L_SCAL = L_SCR + 32*8;                   // double[8] scalars
constexpr unsigned SMEM_BYTES = L_SCAL + 8*8;               // 312504 B

typedef float v2f __attribute__((ext_vector_type(2)));
typedef float v8f __attribute__((ext_vector_type(8)));
typedef unsigned int uv4 __attribute__((ext_vector_type(4)));
typedef int iv4 __attribute__((ext_vector_type(4)));
typedef int iv8 __attribute__((ext_vector_type(8)));

#if defined(__has_builtin)
#  if __has_builtin(__builtin_amdgcn_wmma_f32_16x16x4_f32)
#    define HAVE_WMMA 1
#  endif
#  if __has_builtin(__builtin_amdgcn_tensor_load_to_lds)
#    define HAVE_TDM 1
#  endif
#endif

__device__ __forceinline__ float waveRedF32(float v) {
  #pragma unroll
  for (int o = 16; o > 0; o >>= 1) v += __shfl_xor(v, o, 32);
  return v;
}
__device__ __forceinline__ double waveRedF64(double v) {
  #pragma unroll
  for (int o = 16; o > 0; o >>= 1) v += __shfl_xor(v, o, 32);
  return v;
}
__device__ __forceinline__ constexpr int trimap(int i, int j) { // requires i<=j
  return i*6 + j - (i*(i+1))/2;
}

// ---------------------------------------------------------------------------
// Kernel 0: zero resid/total; build fixed-DOF mask and per-DOF fix values
// ---------------------------------------------------------------------------
__global__ void init_kernel(const int* __restrict__ fixn,
                            const float* __restrict__ fixv,
                            double* __restrict__ resid, double* __restrict__ total,
                            char* __restrict__ mask, double* __restrict__ fvDof) {
  int i = blockIdx.x * blockDim.x + threadIdx.x;
  if (i >= NDOF) return;
  resid[i] = 0.0;
  char m = 0; double fv = 0.0;
  for (int t = 0; t < NFIX; ++t) {
    int w = 2*fixn[2*t] + fixn[2*t+1];
    if (w == i) { m = 1; fv = (double)fixv[t]; }
  }
  mask[i] = m; fvDof[i] = fv;
  if (i == 0) *total = 0.0;
}

// ---------------------------------------------------------------------------
// Kernel 1: per-element stiffness (f64 -> f32 symmetric-triangle SoA),
//           symmetric BC elimination baked in, u16 node table,
//           material output + total accumulation
// ---------------------------------------------------------------------------
__global__ void elem_kernel(const float* __restrict__ coord,
                            const int*   __restrict__ connect,
                            const float* __restrict__ emat,
                            const char*  __restrict__ mask,
                            float* __restrict__ est,              // [NTRI][NELEM]
                            unsigned short* __restrict__ nod,     // [3][NELEM]
                            double* __restrict__ total,
                            float* __restrict__ mat_out) {
  int e = blockIdx.x * blockDim.x + threadIdx.x;
  if (e >= NELEM) return;

  int a = connect[3*e+0], b = connect[3*e+1], c = connect[3*e+2];
  double xa = coord[2*a], ya = coord[2*a+1];
  double xb = coord[2*b], yb = coord[2*b+1];
  double xc = coord[2*c], yc = coord[2*c+1];

  double den_a = (ya-yb)*(xc-xb) - (xa-xb)*(yc-yb);
  double den_b = (yb-yc)*(xa-xc) - (xb-xc)*(ya-yc);
  double den_c = (yc-ya)*(xb-xa) - (xc-xa)*(yb-ya);
  double nax = -(yc-yb)/den_a, nay = (xc-xb)/den_a;
  double nbx = -(ya-yc)/den_b, nby = (xa-xc)/den_b;
  double ncx = -(yb-ya)/den_c, ncy = (xb-xa)/den_c;
  double area = 0.5 * fabs((xb-xa)*(yc-ya) - (xc-xa)*(yb-ya));

  double s = 1.0 / (1.0 + exp(-(double)emat[e]));
  const double fST = 200.0/(1.3*0.4), fAL = 69.0/(1.33*0.34);
  double d11 = s*(fAL*0.67) + (1.0-s)*(fST*0.7);
  double d12 = s*(fAL*0.33) + (1.0-s)*(fST*0.3);
  double d33 = s*(fAL*0.17) + (1.0-s)*(fST*0.2);

  double B[3][6] = {
    { nax, 0.0, nbx, 0.0, ncx, 0.0 },
    { 0.0, nay, 0.0, nby, 0.0, ncy },
    { nay, nax, nby, nbx, ncy, ncx } };
  double D[3][3] = { { d11, d12, 0.0 }, { d12, d11, 0.0 }, { 0.0, 0.0, d33 } };

  double M[3][6];
  #pragma unroll
  for (int i = 0; i < 3; ++i)
    #pragma unroll
    for (int j = 0; j < 6; ++j)
      M[i][j] = D[i][0]*B[0][j] + D[i][1]*B[1][j] + D[i][2]*B[2][j];

  int gdof[6] = { 2*a, 2*a+1, 2*b, 2*b+1, 2*c, 2*c+1 };
  bool fz[6];
  #pragma unroll
  for (int i = 0; i < 6; ++i) fz[i] = (mask[gdof[i]] != 0);

  // Symmetric elimination of fixed DOFs (zero row+col). Identical in effect to
  // the reference's row-only elimination because p[fixed]=r[fixed]=0 exactly
  // (fix_vals==0), so fixed columns never contribute; Kp[fixed]=p[fixed] is
  // enforced separately each iteration.
  #pragma unroll
  for (int i = 0; i < 6; ++i)
    #pragma unroll
    for (int j = i; j < 6; ++j) {
      double v = (fz[i] || fz[j]) ? 0.0
               : area * (B[0][i]*M[0][j] + B[1][i]*M[1][j] + B[2][i]*M[2][j]);
      est[trimap(i,j)*NELEM + e] = (float)v;
    }

  nod[0*NELEM+e] = (unsigned short)a;
  nod[1*NELEM+e] = (unsigned short)b;
  nod[2*NELEM+e] = (unsigned short)c;

  mat_out[e] = (float)s;
  __hip_atomic_fetch_add(total, s, __ATOMIC_RELAXED, __HIP_MEMORY_SCOPE_AGENT);
}

// ---------------------------------------------------------------------------
// Kernel 2: traction loads -> resid (f64 global atomics)
// ---------------------------------------------------------------------------
__global__ void resid_kernel(const float* __restrict__ coord,
                             const int*   __restrict__ connect,
                             const int*   __restrict__ didx,
                             const float* __restrict__ dt,
                             double* __restrict__ resid) {
  int t = threadIdx.x;
  if (t >= NDLOAD) return;
  int ei = didx[2*t], side = didx[2*t+1];
  int ptr = (side == 0) ? 1 : (side == 1 ? 2 : 0);   // pointer = [1,2,0]
  int la = connect[3*ei + side];
  int lb = connect[3*ei + ptr];
  double dx = (double)coord[2*la]   - (double)coord[2*lb];
  double dy = (double)coord[2*la+1] - (double)coord[2*lb+1];
  double half = 0.5 * sqrt(dx*dx + dy*dy);
  double tx = (double)dt[2*t] * half, ty = (double)dt[2*t+1] * half;
  __hip_atomic_fetch_add(&resid[2*la],   tx, __ATOMIC_RELAXED, __HIP_MEMORY_SCOPE_AGENT);
  __hip_atomic_fetch_add(&resid[2*la+1], ty, __ATOMIC_RELAXED, __HIP_MEMORY_SCOPE_AGENT);
  __hip_atomic_fetch_add(&resid[2*lb],   tx, __ATOMIC_RELAXED, __HIP_MEMORY_SCOPE_AGENT);
  __hip_atomic_fetch_add(&resid[2*lb+1], ty, __ATOMIC_RELAXED, __HIP_MEMORY_SCOPE_AGENT);
}

// ---------------------------------------------------------------------------
// TDM: 1-D tile DMA Global -> LDS (tile_dim0 = element count, <= 65535)
// ---------------------------------------------------------------------------
#ifdef HAVE_TDM
__device__ __forceinline__ void tdm_load_1d(unsigned lds_off, const void* gptr,
                                            unsigned nelems, int dsize_code) {
  uv4 g0; iv8 g1; iv4 g2 = (iv4)0; iv4 g3 = (iv4)0;
  unsigned long long ga = (unsigned long long)(size_t)gptr;
  g0.x = 1u;                                               // count=1, user D#
  g0.y = lds_off;                                          // lds_addr
  g0.z = (unsigned)(ga & 0xFFFFFFFFu);                     // global_addr[31:0]
  g0.w = (unsigned)((ga >> 32) & 0x01FFFFFFu) | (2u << 30);// addr[56:32] | type=2
  g1[0] = (int)((unsigned)dsize_code << 16);               // data_size
  g1[1] = (int)((nelems & 0xFFFFu) << 16);                 // tensor_dim0[15:0]
  g1[2] = (int)((nelems >> 16) & 0xFFFFu) | (1 << 16);     // dim0[31:16] | dim1=1
  g1[3] = (int)((nelems & 0xFFFFu) << 16);                 // tile_dim0
  g1[4] = 1;                                               // tile_dim1 = 1
  g1[5] = (int)nelems;                                     // tensor_dim0_stride
  g1[6] = 0; g1[7] = 0;
#if __clang_major__ >= 23
  iv8 g4 = (iv8)0;
  __builtin_amdgcn_tensor_load_to_lds(g0, g1, g2, g3, g4, 0);
#else
  __builtin_amdgcn_tensor_load_to_lds(g0, g1, g2, g3, 0);
#endif
}
#endif

// ---------------------------------------------------------------------------
// Kernel 3: persistent single-workgroup CG (1024 thr = 32 wave32).
// All hot state LDS-resident; u,r live in VGPRs (4 owned slots/thread).
// ---------------------------------------------------------------------------
__global__ void __launch_bounds__(1024, 1)
cg_kernel(const float* __restrict__ coord,
          const double* __restrict__ resid_ws,
          const char*  __restrict__ mask_ws,
          const double* __restrict__ fv_ws,
          const float* __restrict__ est_ws,             // [NTRI][NELEM]
          const unsigned short* __restrict__ nod_ws,    // [3][NELEM]
          const double* __restrict__ total_ws,
          float* __restrict__ out) {
  extern __shared__ __align__(16) char smem[];
  float*          estL = (float*)(smem + L_EST);
  double*         p    = (double*)(smem + L_P);
  float*          Kp   = (float*)(smem + L_KP);
  unsigned short* nodL = (unsigned short*)(smem + L_NOD);
  double*         scr  = (double*)(smem + L_SCR);
  double*         scal = (double*)(smem + L_SCAL);

  const int tid  = threadIdx.x;
  const int lane = tid & 31, wave = tid >> 5;

  // ---- per-thread owned slots: i = tid + 1024k ----
  double u_reg[4] = {0,0,0,0}, r_reg[4] = {0,0,0,0};
  bool   m_reg[4] = {false,false,false,false};

  #pragma unroll
  for (int k = 0; k < 4; ++k) {
    int i = tid + (k << 10);
    if (i < NDOF) {
      bool m = (mask_ws[i] != 0);
      double rv = m ? fv_ws[i] : resid_ws[i];
      m_reg[k] = m; r_reg[k] = rv; p[i] = rv;
    }
  }

  // ---- TDM: DMA stiffness table (255.5 KB) + node table (18.3 KB) -> LDS ----
#ifdef HAVE_TDM
  if (tid == 0) {
    tdm_load_1d((unsigned)(size_t)(void*)estL, est_ws, (unsigned)(NTRI*NELEM), 2);
    tdm_load_1d((unsigned)(size_t)(void*)nodL, nod_ws, (unsigned)(3*NELEM),    1);
#if __has_builtin(__builtin_amdgcn_s_wait_tensorcnt)
    __builtin_amdgcn_s_wait_tensorcnt(0);
#else
    asm volatile("s_wait_tensorcnt 0" ::: "memory");
#endif
  }
#else
  for (int i = tid; i < NTRI*NELEM; i += 1024) estL[i] = est_ws[i];
  for (int i = tid; i < 3*NELEM;    i += 1024) nodL[i] = nod_ws[i];
#endif
  __syncthreads();

  // ---- rr0 (f64, from register-resident r) ----
  {
    double part = 0.0;
    #pragma unroll
    for (int k = 0; k < 4; ++k) part += r_reg[k]*r_reg[k];
    part = waveRedF64(part);
    if (lane == 0) scr[wave] = part;
  }
  __syncthreads();
  if (wave == 0) {
    double v = waveRedF64(scr[lane]);
    if (lane == 0) scal[0] = v;               // rr
  }
  __syncthreads();

  // ---- 3200 CG iterations, matrix-free K*p, all LDS-resident ----
  for (int it = 0; it < NITER; ++it) {
    // phase Z: zero Kp; identity rows for fixed DOFs (Kp[w] = p[w])
    #pragma unroll
    for (int k = 0; k < 4; ++k) {
      int i = tid + (k << 10);
      if (i < NDOF) Kp[i] = m_reg[k] ? (float)p[i] : 0.f;
    }
    __syncthreads();

    // element apply: Kp += A_e * p_e   (est + nodes + p + Kp all in LDS)
    for (int e = tid; e < NELEM; e += 1024) {
      int na = nodL[0*NELEM+e], nb = nodL[1*NELEM+e], nc = nodL[2*NELEM+e];
      int idx[6] = { 2*na, 2*na+1, 2*nb, 2*nb+1, 2*nc, 2*nc+1 };
      float plf[6];
      #pragma unroll
      for (int kk = 0; kk < 6; ++kk) plf[kk] = (float)p[idx[kk]];
      float Es[NTRI];
      #pragma unroll
      for (int t = 0; t < NTRI; ++t) Es[t] = estL[t*NELEM + e];
      #pragma unroll
      for (int i = 0; i < 6; ++i) {
        float acc = 0.f;
        #pragma unroll
        for (int j = 0; j < 6; ++j) {
          int t = (i <= j) ? trimap(i, j) : trimap(j, i);
          acc = fmaf(Es[t], plf[j], acc);
        }
        __hip_atomic_fetch_add(&Kp[idx[i]], acc,
                               __ATOMIC_RELAXED, __HIP_MEMORY_SCOPE_WORKGROUP);
      }
    }
    __syncthreads();

    // p . Kp via V_WMMA_F32_16X16X4_F32 diagonal trick (64 values / instr)
#ifdef HAVE_WMMA
    {
      v8f acc = {};
      for (int c = wave; c < NCHUNK; c += 32) {   // wave-uniform: EXEC all-1s
        int base = (c << 6) + ((lane & 15) << 2) + ((lane >> 4) << 1);
        v2f a, b;
        a.x = (float)p[base];  a.y = (float)p[base+1];
        b.x = Kp[base];        b.y = Kp[base+1];
        acc = __builtin_amdgcn_wmma_f32_16x16x4_f32(false, a, false, b,
                                                    (short)0, acc, false, false);
      }
      float dv = 0.f;
      #pragma unroll
      for (int i = 0; i < 8; ++i) if ((lane & 7) == i) dv = acc[i];
      if (!((lane < 8) || (lane >= 24))) dv = 0.f;
      dv = waveRedF32(dv);
      if (lane == 0) scr[wave] = (double)dv;
    }
#else
    {
      double part = 0.0;
      #pragma unroll
      for (int k = 0; k < 4; ++k) {
        int i = tid + (k << 10);
        if (i < NDOF) part += p[i] * (double)Kp[i];
      }
      part = waveRedF64(part);
      if (lane == 0) scr[wave] = part;
    }
#endif
    __syncthreads();
    if (wave == 0) {
      double v = waveRedF64(scr[lane]);
      if (lane == 0) scal[1] = scal[0] / v;     // alpha = rr / pKp
    }
    __syncthreads();
    double alpha = scal[1];

    // fused axpy: u += a*p ; r -= a*Kp ; accumulate rr_new (f64 registers)
    {
      double part = 0.0;
      #pragma unroll
      for (int k = 0; k < 4; ++k) {
        int i = tid + (k << 10);
        if (i < NDOF) {
          u_reg[k] += alpha * p[i];
          r_reg[k] -= alpha * (double)Kp[i];
          part += r_reg[k] * r_reg[k];
        }
      }
      part = waveRedF64(part);
      if (lane == 0) scr[wave] = part;
    }
    __syncthreads();
    if (wave == 0) {
      double rrn = waveRedF64(scr[lane]);
      if (lane == 0) { scal[2] = rrn / scal[0]; scal[0] = rrn; }  // beta
    }
    __syncthreads();
    double beta = scal[2];

    #pragma unroll
    for (int k = 0; k < 4; ++k) {
      int i = tid + (k << 10);
      if (i < NDOF) p[i] = r_reg[k] + beta * p[i];
    }
    __syncthreads();
  }

  // ---- outputs (f32) from register-resident u ----
  #pragma unroll
  for (int k = 0; k < 4; ++k) {
    int i = tid + (k << 10);
    if (i < NDOF) {
      double uv = u_reg[k];
      out[OU + i] = (float)uv;
      int n = i >> 1;
      if ((i & 1) == 0) out[OUX + n] = (float)((double)coord[2*n]   + uv);
      else              out[OUY + n] = (float)((double)coord[2*n+1] + uv);
    }
  }
  if (tid == 0) out[OTOT] = (float)(*total_ws);
}

// ---------------------------------------------------------------------------
// Host launcher
// ---------------------------------------------------------------------------
extern "C" void kernel_launch(void* const* d_in, const int* in_sizes, int n_in,
                              void* d_out, int out_size, void* d_ws, size_t ws_size,
                              hipStream_t stream) {
  const float* coord   = (const float*)d_in[0];
  const int*   connect = (const int*)  d_in[1];
  const float* emat    = (const float*)d_in[2];
  const int*   fixn    = (const int*)  d_in[3];
  const float* fixv    = (const float*)d_in[4];
  const int*   didx    = (const int*)  d_in[5];
  const float* dt      = (const float*)d_in[6];
  float* out = (float*)d_out;

  char* ws = (char*)d_ws;
  double* resid = (double*)(ws + WS_RESID);
  double* total = (double*)(ws + WS_TOTAL);
  char*   mask  = (char*)  (ws + WS_MASK);
  double* fvDof = (double*)(ws + WS_FV);
  float*  est   = (float*) (ws + WS_EST);
  unsigned short* nod = (unsigned short*)(ws + WS_NOD);

  (void)hipFuncSetAttribute((const void*)cg_kernel,
                            hipFuncAttributeMaxDynamicSharedMemorySize,
                            (int)SMEM_BYTES);

  init_kernel <<<(NDOF  + 255)/256, 256, 0, stream>>>(fixn, fixv, resid, total,
                                                      mask, fvDof);
  elem_kernel <<<(NELEM + 255)/256, 256, 0, stream>>>(coord, connect, emat, mask,
                                                      est, nod, total, out + OMAT);
  resid_kernel<<<1, 64, 0, stream>>>(coord, connect, didx, dt, resid);
  cg_kernel   <<<1, 1024, SMEM_BYTES, stream>>>(coord, resid, mask, fvDof,
                                                est, nod, total, out);
}